// LinearAttention3d_32384053412236
// MI455X (gfx1250) — compile-verified
//
#include <hip/hip_runtime.h>
#include <hip/hip_bf16.h>

typedef __attribute__((ext_vector_type(2))) float v2f;
typedef __attribute__((ext_vector_type(4))) float v4f;
typedef __attribute__((ext_vector_type(8))) float v8f;

#define BATCH    4
#define CCH      128
#define HEADS    8
#define DIM_HEAD 32
#define HIDDEN   256      // HEADS * DIM_HEAD
#define NSEQ     32768    // 32*32*32
#define QKV_ROWS 768      // 3 * HIDDEN
#define SCALE_Q  0.17677669529663687f  // 32^-0.5

// LDS layout for the B tile: k-pair-interleaved, padded.
// sB[p * PAIR_STRIDE + n*2 + (k&1)]  with p = k>>1.
// Pad of 32 floats (=32 banks) keeps the two half-waves (adjacent k-pairs)
// on disjoint halves of the 64-bank LDS -> conflict-free ds_load_b64 frags.
#define PAIR_STRIDE 288   // 128*2 + 32

static __device__ inline v8f vzero8() {
  v8f z;
#pragma unroll
  for (int i = 0; i < 8; ++i) z[i] = 0.0f;
  return z;
}

// ---------------------------------------------------------------------------
// Generic FP32 WMMA GEMM:  C[b] = A[b] (MxK, row stride lda) * B[b] (KxN) + bias
// Block tile 128x128, 8 waves x (16 M x 128 N) strips, V_WMMA_F32_16X16X4_F32.
// Register-staged double-buffered LDS for B; K must be a multiple of 32.
// ---------------------------------------------------------------------------
__global__ __launch_bounds__(256) void gemm_wmma_f32(
    const float* __restrict__ A, long long aStride, int lda,
    const float* __restrict__ B, long long bStride,
    const float* __restrict__ bias,
    float* __restrict__ C, long long cStride, int K) {
  __shared__ float sB[2][16 * PAIR_STRIDE];
  const int N = NSEQ;
  const int bz = blockIdx.z;
  const float* Ab = A + aStride * bz;
  const float* Bb = B + bStride * bz;
  float* Cb = C + cStride * bz;
  const int n0   = blockIdx.x * 128;
  const int m0   = blockIdx.y * 128;
  const int tid  = threadIdx.x;
  const int wave = tid >> 5;
  const int lane = tid & 31;
  const int half = lane >> 4;   // which 16-lane half of the wave
  const int l15  = lane & 15;
  // A fragment base: row = m0 + wave*16 + lane%16, k-pair offset 2*half
  const long long arow = (long long)(m0 + wave * 16 + l15) * lda + 2 * half;

  v8f acc[8];
#pragma unroll
  for (int i = 0; i < 8; ++i) acc[i] = vzero8();

  // Each thread stages 4 float4s per 32x128 chunk:
  // item i -> k-pair p = i>>6, n-pair np = i&63 (n = 2*np).
  v4f stage[4];

  auto loadChunk = [&](int kc) {
#pragma unroll
    for (int j = 0; j < 4; ++j) {
      const int i = tid + 256 * j;
      const int p = i >> 6;
      const int np = i & 63;
      const float* s0 = Bb + (long long)(kc + 2 * p) * N + n0 + np * 2;
      const v2f lo = *(const v2f*)s0;        // row k0, cols n..n+1
      const v2f hi = *(const v2f*)(s0 + N);  // row k1, cols n..n+1
      v4f v;
      v.x = lo.x; v.y = hi.x; v.z = lo.y; v.w = hi.y;  // interleave (n,k)
      stage[j] = v;
    }
  };
  auto storeChunk = [&](int buf) {
#pragma unroll
    for (int j = 0; j < 4; ++j) {
      const int i = tid + 256 * j;
      const int p = i >> 6;
      const int np = i & 63;
      *(v4f*)&sB[buf][p * PAIR_STRIDE + np * 4] = stage[j];
    }
  };

  const int nchunks = K / 32;
  loadChunk(0);
  storeChunk(0);
  __syncthreads();

  for (int c = 0; c < nchunks; ++c) {
    const int kc = c * 32;
    const int buf = c & 1;
    if (c + 1 < nchunks) loadChunk(kc + 32);  // overlap with compute below
    if (c + 2 < nchunks) {
      // prefetch chunk-after-next (global_prefetch_b8); 256 addrs cover 16KB
      __builtin_prefetch(Bb + (long long)(kc + 64 + (tid >> 3)) * N + n0 +
                             (tid & 7) * 16, 0, 0);
    }
    // batch-load all A fragments for this 32-deep chunk
    v2f afrag[8];
#pragma unroll
    for (int k4 = 0; k4 < 8; ++k4) {
      afrag[k4] = *(const v2f*)(Ab + arow + kc + k4 * 4);
    }
#pragma unroll
    for (int k4 = 0; k4 < 8; ++k4) {
      const int prow = k4 * 2 + half;  // (k4*4 + 2*half) >> 1
      const float* bbase = &sB[buf][prow * PAIR_STRIDE + l15 * 2];
#pragma unroll
      for (int nt = 0; nt < 8; ++nt) {
        const v2f bf = *(const v2f*)(bbase + nt * 32);  // one ds_load_b64
        acc[nt] = __builtin_amdgcn_wmma_f32_16x16x4_f32(
            false, afrag[k4], false, bf, (short)0, acc[nt], false, false);
      }
    }
    __syncthreads();
    if (c + 1 < nchunks) {
      storeChunk(buf ^ 1);
      __syncthreads();
    }
  }

  // store: acc[nt][r] -> C[m0 + wave*16 + r + 8*half][n0 + nt*16 + l15]
#pragma unroll
  for (int nt = 0; nt < 8; ++nt) {
#pragma unroll
    for (int r = 0; r < 8; ++r) {
      const int m = m0 + wave * 16 + r + 8 * half;
      const int n = n0 + nt * 16 + l15;
      float v = acc[nt][r];
      if (bias) v += bias[m];
      Cb[(long long)m * N + n] = v;
    }
  }
}

// ---------------------------------------------------------------------------
// q softmax over the feature axis (32 channels within a head), * SCALE.
// ---------------------------------------------------------------------------
__global__ __launch_bounds__(256) void qsoftmax_kernel(float* __restrict__ qkv) {
  const long long n = (long long)blockIdx.x * 256 + threadIdx.x;
  const int h = blockIdx.y;
  const int b = blockIdx.z;
  float* base = qkv + ((long long)b * QKV_ROWS + h * DIM_HEAD) * NSEQ + n;
  float vals[DIM_HEAD];
  float m = -3.0e38f;
#pragma unroll
  for (int d = 0; d < DIM_HEAD; ++d) {
    vals[d] = base[(long long)d * NSEQ];
    m = fmaxf(m, vals[d]);
  }
  float s = 0.0f;
#pragma unroll
  for (int d = 0; d < DIM_HEAD; ++d) {
    vals[d] = __expf(vals[d] - m);
    s += vals[d];
  }
  const float inv = SCALE_Q / s;
#pragma unroll
  for (int d = 0; d < DIM_HEAD; ++d) base[(long long)d * NSEQ] = vals[d] * inv;
}

// ---------------------------------------------------------------------------
// k: per-row max, write exp(x - max) in place, emit row sum (normalization
// folded into W2). One block per row.
// ---------------------------------------------------------------------------
__global__ __launch_bounds__(256) void kexp_kernel(float* __restrict__ qkv,
                                                   float* __restrict__ ksum) {
  const int row = blockIdx.x;          // 0..BATCH*HIDDEN-1
  const int b = row >> 8;
  const int c = row & 255;
  float* p = qkv + ((long long)b * QKV_ROWS + HIDDEN + c) * NSEQ;
  const int tid = threadIdx.x;
  __shared__ float red[256];

  float m = -3.0e38f;
  for (int i = tid; i < NSEQ; i += 256) m = fmaxf(m, p[i]);
  red[tid] = m;
  __syncthreads();
  for (int s = 128; s > 0; s >>= 1) {
    if (tid < s) red[tid] = fmaxf(red[tid], red[tid + s]);
    __syncthreads();
  }
  m = red[0];
  __syncthreads();

  float sum = 0.0f;
  for (int i = tid; i < NSEQ; i += 256) {
    const float e = __expf(p[i] - m);
    p[i] = e;
    sum += e;
  }
  red[tid] = sum;
  __syncthreads();
  for (int s = 128; s > 0; s >>= 1) {
    if (tid < s) red[tid] += red[tid + s];
    __syncthreads();
  }
  if (tid == 0) ksum[row] = red[0];
}

// ---------------------------------------------------------------------------
// ctx[b,h,d,e] += sum_n expk[b,h,d,n] * v[b,h,e,n]  via WMMA, split-K over n.
// ---------------------------------------------------------------------------
__global__ __launch_bounds__(256) void ctx_kernel(const float* __restrict__ qkv,
                                                  float* __restrict__ ctx) {
  const int bh = blockIdx.z;        // b*8 + h
  const int b = bh >> 3, h = bh & 7;
  const int tile = blockIdx.y;      // 2x2 tiling of 32x32
  const int d0 = (tile >> 1) * 16;
  const int e0 = (tile & 1) * 16;
  const int lane = threadIdx.x & 31;
  const int wave = threadIdx.x >> 5;
  const int split = blockIdx.x * 8 + wave;  // 0..63
  const int half = lane >> 4;
  const int l15 = lane & 15;

  const float* kp = qkv + ((long long)b * QKV_ROWS + HIDDEN + h * DIM_HEAD + d0) * NSEQ;
  const float* vp = qkv + ((long long)b * QKV_ROWS + 2 * HIDDEN + h * DIM_HEAD + e0) * NSEQ;

  v8f acc = vzero8();
  const long long nbase = (long long)split * 512;
  for (int k = 0; k < 512; k += 4) {
    const long long n = nbase + k + 2 * half;
    v2f a = *(const v2f*)(kp + (long long)l15 * NSEQ + n);   // expk[d, n]
    v2f bf = *(const v2f*)(vp + (long long)l15 * NSEQ + n);  // v^T[n, e]
    acc = __builtin_amdgcn_wmma_f32_16x16x4_f32(
        false, a, false, bf, (short)0, acc, false, false);
  }
  float* cbase = ctx + (long long)bh * DIM_HEAD * DIM_HEAD;
#pragma unroll
  for (int r = 0; r < 8; ++r) {
    atomicAdd(cbase + (d0 + r + 8 * half) * DIM_HEAD + e0 + l15, acc[r]);
  }
}

// ---------------------------------------------------------------------------
// W2[b][o][h*32+d] = (1/ksum[b][h*32+d]) * sum_e w_out[o][h*32+e]*ctx[b,h,d,e]
// ---------------------------------------------------------------------------
__global__ __launch_bounds__(256) void w2_kernel(const float* __restrict__ w_out,
                                                 const float* __restrict__ ctx,
                                                 const float* __restrict__ ksum,
                                                 float* __restrict__ W2) {
  const int idx = blockIdx.x * 256 + threadIdx.x;  // b*128*256 + o*256 + hd
  const int b = idx >> 15;
  const int o = (idx >> 8) & 127;
  const int hd = idx & 255;
  const int h = hd >> 5, d = hd & 31;
  const float* cc = ctx + (((long long)b * HEADS + h) * DIM_HEAD + d) * DIM_HEAD;
  const float* wo = w_out + o * HIDDEN + h * DIM_HEAD;
  float s = 0.0f;
#pragma unroll
  for (int e = 0; e < DIM_HEAD; ++e) s += wo[e] * cc[e];
  W2[idx] = s / ksum[b * HIDDEN + hd];
}

// ---------------------------------------------------------------------------
extern "C" void kernel_launch(void* const* d_in, const int* in_sizes, int n_in,
                              void* d_out, int out_size, void* d_ws, size_t ws_size,
                              hipStream_t stream) {
  const float* x     = (const float*)d_in[0];  // [4,128,32,32,32]
  const float* w_qkv = (const float*)d_in[1];  // [768,128]
  const float* w_out = (const float*)d_in[2];  // [128,256]
  const float* b_out = (const float*)d_in[3];  // [128]
  float* out = (float*)d_out;                  // [4,128,32,32,32]

  float* qkv = (float*)d_ws;                              // [4,768,32768]
  const size_t qkvElems = (size_t)BATCH * QKV_ROWS * NSEQ;
  float* ksum = qkv + qkvElems;                           // [4*256]
  float* ctx  = ksum + BATCH * HIDDEN;                    // [4,8,32,32]
  float* W2   = ctx + (size_t)BATCH * HEADS * DIM_HEAD * DIM_HEAD;  // [4,128,256]

  const dim3 blk(256);

  // 1) qkv[b] = w_qkv @ x[b]
  gemm_wmma_f32<<<dim3(NSEQ / 128, QKV_ROWS / 128, BATCH), blk, 0, stream>>>(
      w_qkv, 0ll, CCH, x, (long long)CCH * NSEQ, nullptr,
      qkv, (long long)QKV_ROWS * NSEQ, CCH);

  // 2) q softmax over feature dim (in place, * SCALE)
  qsoftmax_kernel<<<dim3(NSEQ / 256, HEADS, BATCH), blk, 0, stream>>>(qkv);

  // 3) k -> exp(k - rowmax) in place, row sums to ksum
  kexp_kernel<<<dim3(BATCH * HIDDEN), blk, 0, stream>>>(qkv, ksum);

  // 4) ctx = expk @ v^T  (split-K + atomics)
  hipMemsetAsync(ctx, 0, sizeof(float) * BATCH * HEADS * DIM_HEAD * DIM_HEAD, stream);
  ctx_kernel<<<dim3(8, 4, BATCH * HEADS), blk, 0, stream>>>(qkv, ctx);

  // 5) W2 = w_out @ ctx (per head), normalized by ksum
  w2_kernel<<<dim3((BATCH * CCH * HIDDEN) / 256), blk, 0, stream>>>(w_out, ctx, ksum, W2);

  // 6) out[b] = W2[b] @ q_softmaxed[b] + b_out
  gemm_wmma_f32<<<dim3(NSEQ / 128, 1, BATCH), blk, 0, stream>>>(
      W2, (long long)CCH * HIDDEN, HIDDEN, qkv, (long long)QKV_ROWS * NSEQ, b_out,
      out, (long long)CCH * NSEQ, HIDDEN);
}